// RegressionLayer_11699490915134
// MI455X (gfx1250) — compile-verified
//
#include <hip/hip_runtime.h>
#include <math.h>

// ---------------------------------------------------------------------------
// MI455X (gfx1250) implementation of the conv + GCN + edge-attention layer.
// Dense GEMM parts use V_WMMA_F32_16X16X4_F32 (fp32 WMMA, wave32).
// Edge gather/scatter phases are wave-per-edge; all node tensors (33.5 MB each)
// are L2-resident on the 192 MB L2, so scattered access stays on-chip.
// ---------------------------------------------------------------------------

typedef __attribute__((ext_vector_type(2))) float v2f;
typedef __attribute__((ext_vector_type(8))) float v8f;

#define CIN   32
#define CMID  64
#define HH    256
#define WW    256
#define EPS   1e-5f
#define ELU_A 0.1f

__device__ __forceinline__ v8f wmma4(v2f a, v2f b, v8f c) {
  // D = A(16x4) * B(4x16) + C(16x16), fp32. 8-arg form:
  // (neg_a, A, neg_b, B, c_mod, C, reuse_a, reuse_b)
  return __builtin_amdgcn_wmma_f32_16x16x4_f32(false, a, false, b, (short)0, c,
                                               false, false);
}

__device__ __forceinline__ float elu_f(float x) {
  return x > 0.f ? x : ELU_A * expm1f(x);
}

// ---------------------------------------------------------------------------
__global__ void k_fill(float* p, float val, int n) {
  int i = blockIdx.x * blockDim.x + threadIdx.x;
  if (i < n) p[i] = val;
}

// Repack conv weights (co, ci, ky, kx) -> Wt[tap][co][ci] so B-operand loads
// in the conv kernel are contiguous 8-byte vector loads.
__global__ void k_wprep(const float* __restrict__ cw, float* __restrict__ wt) {
  int i = blockIdx.x * blockDim.x + threadIdx.x;   // 9*64*32 threads
  if (i >= 9 * CMID * CIN) return;
  int ci  = i & (CIN - 1);
  int co  = (i >> 5) & (CMID - 1);
  int tap = i >> 11;
  wt[i] = cw[((size_t)co * CIN + ci) * 9 + tap];
}

// ---------------------------------------------------------------------------
// 3x3 conv (CIN=32 -> CMID=64) + BN + ELU, implicit GEMM.
// One wave computes a 16-pixel horizontal strip x all 64 output channels.
// A (16 px x 4 ci), B (4 ci x 16 co); lane l: li = l&15, hi = l>>4,
// operand VGPR j holds K = k0 + 2*hi + j (per ISA 16x4 f32 layout).
// Halo handling: clamp coordinates (loads always in-bounds, unconditional)
// and multiply by a 0/1 mask -> no EXEC branching around loads.
__global__ void __launch_bounds__(256)
k_conv_bn_elu(const float* __restrict__ xin, const float* __restrict__ wt,
              const float* __restrict__ g, const float* __restrict__ bb,
              const float* __restrict__ mm, const float* __restrict__ vv,
              float* __restrict__ h0) {
  int wave = (blockIdx.x * blockDim.x + threadIdx.x) >> 5;
  int lane = threadIdx.x & 31;
  int li = lane & 15, hi = lane >> 4;

  int tx = wave & (WW / 16 - 1);     // 16 strips per row
  int gy = wave >> 4;                // b*HH + y
  int bb_ = gy >> 8;                 // /HH
  int y   = gy & (HH - 1);
  int x0  = tx * 16;

  v8f acc[4] = {};                   // 4 tiles of 16 output channels

#pragma unroll
  for (int tap = 0; tap < 9; ++tap) {
    const int dy = tap / 3 - 1, dx = tap % 3 - 1;
    int yy = y + dy;
    int xx = x0 + li + dx;
    bool ok  = (yy >= 0) && (yy < HH) && (xx >= 0) && (xx < WW);
    float msk = ok ? 1.f : 0.f;
    int yyc = min(max(yy, 0), HH - 1);
    int xxc = min(max(xx, 0), WW - 1);
    const float* arow = xin + ((size_t)((bb_ * HH + yyc) * WW + xxc)) * CIN;
    const float* wrow = wt + (size_t)tap * CMID * CIN;
#pragma unroll
    for (int k0 = 0; k0 < CIN; k0 += 4) {
      int ka = k0 + 2 * hi;
      v2f a = *(const v2f*)(arow + ka);     // unconditional b64 load
      a.x *= msk;
      a.y *= msk;
#pragma unroll
      for (int ot = 0; ot < 4; ++ot) {
        int co = ot * 16 + li;
        v2f b = *(const v2f*)(wrow + (size_t)co * CIN + ka);  // b64 load
        acc[ot] = wmma4(a, b, acc[ot]);
      }
    }
  }

  // Epilogue: BN + ELU. D layout: VGPR i -> pixel M = i + 8*hi, channel = co.
#pragma unroll
  for (int ot = 0; ot < 4; ++ot) {
    int co = ot * 16 + li;
    float sc = g[co] * rsqrtf(vv[co] + EPS);
    float sh = bb[co] - mm[co] * sc;
#pragma unroll
    for (int i = 0; i < 8; ++i) {
      int px   = x0 + i + 8 * hi;
      int node = (bb_ * HH + y) * WW + px;
      float t  = acc[ot][i] * sc + sh;
      h0[(size_t)node * CMID + co] = elu_f(t);
    }
  }
}

// ---------------------------------------------------------------------------
// out[n, co] = sum_c hin[n,c] * W[co,c] (+ bias). One wave = 16 nodes x 16 co.
__global__ void __launch_bounds__(256)
k_linear64(const float* __restrict__ hin, const float* __restrict__ W,
           const float* __restrict__ bias, float* __restrict__ out) {
  int gw   = (blockIdx.x * blockDim.x + threadIdx.x) >> 5;
  int lane = threadIdx.x & 31;
  int li = lane & 15, hi = lane >> 4;
  int ot = gw & 3;
  int nt = gw >> 2;

  const float* hr = hin + ((size_t)(nt * 16 + li)) * CMID;
  const float* wr = W   + ((size_t)(ot * 16 + li)) * CMID;

  v8f acc = {};
#pragma unroll
  for (int k0 = 0; k0 < CMID; k0 += 4) {
    int ka = k0 + 2 * hi;
    v2f a = *(const v2f*)(hr + ka);
    v2f b = *(const v2f*)(wr + ka);
    acc = wmma4(a, b, acc);
  }

  int co = ot * 16 + li;
  float bi = bias ? bias[co] : 0.f;
#pragma unroll
  for (int i = 0; i < 8; ++i) {
    int r = nt * 16 + i + 8 * hi;
    out[(size_t)r * CMID + co] = acc[i] + bi;
  }
}

// ---------------------------------------------------------------------------
__global__ void k_deg(const int* __restrict__ col, const float* __restrict__ ew2,
                      float* __restrict__ deg, int E) {
  int e = blockIdx.x * blockDim.x + threadIdx.x;
  if (e < E) atomicAdd(&deg[col[e]], ew2[(size_t)e * 2 + 1]);
}

__global__ void k_dinv(const float* __restrict__ deg, float* __restrict__ dinv, int n) {
  int i = blockIdx.x * blockDim.x + threadIdx.x;
  if (i < n) { float d = deg[i]; dinv[i] = d > 0.f ? rsqrtf(d) : 0.f; }
}

// GCN aggregate over real edges (self loops folded into k_gcn_post).
// One wave per edge; lane handles channels {l, l+32}.
__global__ void __launch_bounds__(256)
k_gcn_agg(const int* __restrict__ row, const int* __restrict__ col,
          const float* __restrict__ ew2, const float* __restrict__ dinv,
          const float* __restrict__ hw, float* __restrict__ agg) {
  int e    = (blockIdx.x * blockDim.x + threadIdx.x) >> 5;
  int lane = threadIdx.x & 31;
  int r = row[e], c = col[e];
  float norm = dinv[r] * ew2[(size_t)e * 2 + 1] * dinv[c];
  const float* src = hw  + (size_t)r * CMID;
  float*       dst = agg + (size_t)c * CMID;
  atomicAdd(&dst[lane],      norm * src[lane]);
  atomicAdd(&dst[lane + 32], norm * src[lane + 32]);
}

// h1 = elu(bn1(agg + 2*dinv^2*hw + gcn_b))
__global__ void k_gcn_post(const float* __restrict__ agg, const float* __restrict__ hw,
                           const float* __restrict__ dinv, const float* __restrict__ gb,
                           const float* __restrict__ g, const float* __restrict__ b,
                           const float* __restrict__ m, const float* __restrict__ v,
                           float* __restrict__ h1) {
  int i  = blockIdx.x * blockDim.x + threadIdx.x;  // n*64 + ch
  int ch = i & (CMID - 1);
  int n  = i >> 6;
  float di  = dinv[n];
  float val = agg[i] + 2.0f * di * di * hw[i] + gb[ch];
  float sc  = g[ch] * rsqrtf(v[ch] + EPS);
  h1[i] = elu_f((val - m[ch]) * sc + b[ch]);
}

// ---------------------------------------------------------------------------
// alpha[e] = dot(q[col], k[row] + e_vec) / 8 ; segment max into amax[col].
__global__ void __launch_bounds__(256)
k_alpha(const int* __restrict__ row, const int* __restrict__ col,
        const float* __restrict__ ew2, const float* __restrict__ we,
        const float* __restrict__ q, const float* __restrict__ kk,
        float* __restrict__ alpha, float* __restrict__ amax) {
  int e    = (blockIdx.x * blockDim.x + threadIdx.x) >> 5;
  int lane = threadIdx.x & 31;
  int r = row[e], c = col[e];
  float w0 = ew2[(size_t)e * 2 + 0], w1 = ew2[(size_t)e * 2 + 1];
  const float* qr = q  + (size_t)c * CMID;
  const float* kr = kk + (size_t)r * CMID;
  float s = 0.f;
  {
    int ch = lane;
    float ev = we[ch * 2] * w0 + we[ch * 2 + 1] * w1;
    s += qr[ch] * (kr[ch] + ev);
  }
  {
    int ch = lane + 32;
    float ev = we[ch * 2] * w0 + we[ch * 2 + 1] * w1;
    s += qr[ch] * (kr[ch] + ev);
  }
  for (int off = 16; off > 0; off >>= 1) s += __shfl_xor(s, off, 32);
  float a = s * 0.125f;   // / sqrt(64)
  if (lane == 0) { alpha[e] = a; atomicMax(&amax[c], a); }
}

__global__ void k_amaxfix(float* __restrict__ amax, int n) {
  int i = blockIdx.x * blockDim.x + threadIdx.x;
  if (i < n) { float a = amax[i]; amax[i] = __builtin_isfinite(a) ? a : 0.f; }
}

// ex = exp(alpha - amax[col]); denom[col] += ex; acc[col] += ex*(v[row]+e_vec)
__global__ void __launch_bounds__(256)
k_attn_scatter(const int* __restrict__ row, const int* __restrict__ col,
               const float* __restrict__ ew2, const float* __restrict__ we,
               const float* __restrict__ vv, const float* __restrict__ alpha,
               const float* __restrict__ amax, float* __restrict__ denom,
               float* __restrict__ acc) {
  int e    = (blockIdx.x * blockDim.x + threadIdx.x) >> 5;
  int lane = threadIdx.x & 31;
  int r = row[e], c = col[e];
  float ex = expf(alpha[e] - amax[c]);
  if (lane == 0) atomicAdd(&denom[c], ex);
  float w0 = ew2[(size_t)e * 2 + 0], w1 = ew2[(size_t)e * 2 + 1];
  const float* vr  = vv  + (size_t)r * CMID;
  float*       dst = acc + (size_t)c * CMID;
  {
    int ch = lane;
    float ev = we[ch * 2] * w0 + we[ch * 2 + 1] * w1;
    atomicAdd(&dst[ch], ex * (vr[ch] + ev));
  }
  {
    int ch = lane + 32;
    float ev = we[ch * 2] * w0 + we[ch * 2 + 1] * w1;
    atomicAdd(&dst[ch], ex * (vr[ch] + ev));
  }
}

// out[n] = dot(bn1(acc/denom + skip), wl) + bl
__global__ void k_final(const float* __restrict__ acc, const float* __restrict__ denom,
                        const float* __restrict__ skip, const float* __restrict__ g,
                        const float* __restrict__ b, const float* __restrict__ m,
                        const float* __restrict__ v, const float* __restrict__ wl,
                        const float* __restrict__ bl, float* __restrict__ out, int n) {
  int i = blockIdx.x * blockDim.x + threadIdx.x;
  if (i >= n) return;
  float d   = denom[i];
  float inv = d > 0.f ? 1.0f / d : 0.f;
  const float* ar = acc  + (size_t)i * CMID;
  const float* sr = skip + (size_t)i * CMID;
  float s = 0.f;
#pragma unroll
  for (int ch = 0; ch < CMID; ++ch) {
    float val = ar[ch] * inv + sr[ch];
    float sc  = g[ch] * rsqrtf(v[ch] + EPS);
    val = (val - m[ch]) * sc + b[ch];
    s += val * wl[ch];
  }
  out[i] = s + bl[0];
}

// ---------------------------------------------------------------------------
extern "C" void kernel_launch(void* const* d_in, const int* in_sizes, int n_in,
                              void* d_out, int out_size, void* d_ws, size_t ws_size,
                              hipStream_t stream) {
  const float* xin    = (const float*)d_in[0];
  const int*   ei     = (const int*)  d_in[1];
  const float* ew     = (const float*)d_in[2];
  const float* conv_w = (const float*)d_in[6];
  const float* bn2_g  = (const float*)d_in[7];
  const float* bn2_b  = (const float*)d_in[8];
  const float* bn2_m  = (const float*)d_in[9];
  const float* bn2_v  = (const float*)d_in[10];
  const float* gcn_w  = (const float*)d_in[11];
  const float* gcn_b  = (const float*)d_in[12];
  const float* bn1_g  = (const float*)d_in[13];
  const float* bn1_b  = (const float*)d_in[14];
  const float* bn1_m  = (const float*)d_in[15];
  const float* bn1_v  = (const float*)d_in[16];
  const float* wq     = (const float*)d_in[17];
  const float* bq     = (const float*)d_in[18];
  const float* wk     = (const float*)d_in[19];
  const float* bk     = (const float*)d_in[20];
  const float* wv     = (const float*)d_in[21];
  const float* bv     = (const float*)d_in[22];
  const float* we     = (const float*)d_in[23];
  const float* wskip  = (const float*)d_in[24];
  const float* bskip  = (const float*)d_in[25];
  const float* wl     = (const float*)d_in[26];
  const float* bl     = (const float*)d_in[27];

  const int Nn = in_sizes[0] / CIN;   // 131072
  const int E  = in_sizes[1] / 2;     // 1048576
  const int* row = ei;
  const int* col = ei + E;

  float* ws = (float*)d_ws;
  const size_t F = (size_t)Nn * CMID;
  float* R0    = ws;            // h0 -> h1 -> attn accumulator
  float* R1    = ws + F;        // hw -> k
  float* R2    = ws + 2 * F;    // gcn agg -> q
  float* R3    = ws + 3 * F;    // v
  float* R4    = ws + 4 * F;    // skip
  float* DEG   = ws + 5 * F;
  float* DINV  = DEG + Nn;
  float* AMAX  = DINV + Nn;
  float* DEN   = AMAX + Nn;
  float* ALPHA = DEN + Nn;      // E floats
  float* WT    = ALPHA + E;     // 9*64*32 floats (repacked conv weights)

  const int T = 256;
  auto cdiv = [](int a, int b) { return (a + b - 1) / b; };

  // init + weight repack
  k_fill<<<cdiv(Nn, T), T, 0, stream>>>(DEG, 2.0f, Nn);           // self-loop weight
  k_fill<<<cdiv(Nn, T), T, 0, stream>>>(AMAX, -INFINITY, Nn);
  k_fill<<<cdiv(Nn, T), T, 0, stream>>>(DEN, 0.0f, Nn);
  k_fill<<<cdiv((int)F, T), T, 0, stream>>>(R2, 0.0f, (int)F);
  k_wprep<<<cdiv(9 * CMID * CIN, T), T, 0, stream>>>(conv_w, WT);

  // conv + bn2 + elu: one wave per 16-pixel strip -> Nn/16 waves, 8 waves/block
  k_conv_bn_elu<<<Nn / 16 / 8, T, 0, stream>>>(xin, WT, bn2_g, bn2_b, bn2_m,
                                               bn2_v, R0);

  // degree / dinv
  k_deg<<<cdiv(E, T), T, 0, stream>>>(col, ew, DEG, E);
  k_dinv<<<cdiv(Nn, T), T, 0, stream>>>(DEG, DINV, Nn);

  // hw = h @ gcn_w.T : (Nn/16)*4 waves, 8 waves/block
  k_linear64<<<Nn / 32, T, 0, stream>>>(R0, gcn_w, nullptr, R1);

  // edge aggregate + post (bias, bn1, elu, self-loop term)
  k_gcn_agg<<<E / 8, T, 0, stream>>>(row, col, ew, DINV, R1, R2);
  k_gcn_post<<<cdiv((int)F, T), T, 0, stream>>>(R2, R1, DINV, gcn_b, bn1_g, bn1_b,
                                                bn1_m, bn1_v, R0);

  // q / k / v / skip
  k_linear64<<<Nn / 32, T, 0, stream>>>(R0, wq, bq, R2);
  k_linear64<<<Nn / 32, T, 0, stream>>>(R0, wk, bk, R1);
  k_linear64<<<Nn / 32, T, 0, stream>>>(R0, wv, bv, R3);
  k_linear64<<<Nn / 32, T, 0, stream>>>(R0, wskip, bskip, R4);

  // h1 dead now: reuse R0 as the attention message accumulator
  k_fill<<<cdiv((int)F, T), T, 0, stream>>>(R0, 0.0f, (int)F);

  // attention
  k_alpha<<<E / 8, T, 0, stream>>>(row, col, ew, we, R2, R1, ALPHA, AMAX);
  k_amaxfix<<<cdiv(Nn, T), T, 0, stream>>>(AMAX, Nn);
  k_attn_scatter<<<E / 8, T, 0, stream>>>(row, col, ew, we, R3, ALPHA, AMAX, DEN, R0);

  // out = bn1(acc/denom + skip) @ wl.T + bl
  k_final<<<cdiv(Nn, T), T, 0, stream>>>(R0, DEN, R4, bn1_g, bn1_b, bn1_m, bn1_v,
                                         wl, bl, (float*)d_out, Nn);
}